// UUIIModel_36936718745996
// MI455X (gfx1250) — compile-verified
//
#include <hip/hip_runtime.h>
#include <hip/hip_bf16.h>

namespace {

constexpr int U_N  = 100000;
constexpr int I_N  = 50000;
constexpr int K_DIM = 64;
constexpr float EPSF = 1e-12f;

typedef __attribute__((ext_vector_type(2))) float v2f;
typedef __attribute__((ext_vector_type(8))) float v8f;

// ---------------------------------------------------------------------------
// Zero a buffer (float4 vectorized, grid-stride). All our sizes are /4.
// ---------------------------------------------------------------------------
__global__ void zero4_kernel(float4* __restrict__ p, int n4) {
  int i = blockIdx.x * blockDim.x + threadIdx.x;
  int stride = gridDim.x * blockDim.x;
  for (; i < n4; i += stride) p[i] = make_float4(0.f, 0.f, 0.f, 0.f);
}

// ---------------------------------------------------------------------------
// bufA = concat(Gu, Gi); acc (= d_out) = same values (all_emb[0] term of mean)
// ---------------------------------------------------------------------------
__global__ void init_concat_kernel(const float* __restrict__ Gu,
                                   const float* __restrict__ Gi,
                                   float* __restrict__ bufA,
                                   float* __restrict__ acc) {
  int idx = blockIdx.x * blockDim.x + threadIdx.x;
  constexpr int n = (U_N + I_N) * K_DIM;
  if (idx >= n) return;
  float v = (idx < U_N * K_DIM) ? Gu[idx] : Gi[idx - U_N * K_DIM];
  bufA[idx] = v;
  acc[idx]  = v;
}

// ---------------------------------------------------------------------------
// Scatter-add SpMM: y[dst[e]] += w[e] * x[src[e]]   (rows of 64 f32)
// 16 threads per edge, float4 per thread: coalesced 256B gather per edge.
// Relaxed, agent-scope atomic fadd -> native global_atomic_add_f32 (no CAS
// loop, no return value -> STOREcnt path), served by L2 atomic units (the
// hot spmm working set fits in the 192MB L2).
// ---------------------------------------------------------------------------
__device__ __forceinline__ void atomic_fadd_agent(float* p, float v) {
  __hip_atomic_fetch_add(p, v, __ATOMIC_RELAXED, __HIP_MEMORY_SCOPE_AGENT);
}

__global__ void spmm_kernel(const int* __restrict__ src, const int* __restrict__ dst,
                            const float* __restrict__ w, const float* __restrict__ x,
                            float* __restrict__ y, int nE) {
  int tid = blockIdx.x * blockDim.x + threadIdx.x;
  int e = tid >> 4;
  if (e >= nE) return;
  int q = (tid & 15) << 2;
  float we = w[e];
  const float4 xv = *(const float4*)(x + (size_t)src[e] * K_DIM + q);
  float* yp = y + (size_t)dst[e] * K_DIM + q;
  atomic_fadd_agent(yp + 0, we * xv.x);
  atomic_fadd_agent(yp + 1, we * xv.y);
  atomic_fadd_agent(yp + 2, we * xv.z);
  atomic_fadd_agent(yp + 3, we * xv.w);
}

// ---------------------------------------------------------------------------
// Exact fp32 row-norms of a 16x64 tile via V_WMMA_F32_16X16X4_F32.
// On CDNA5, a 16x4 f32 A-tile and its transpose (as B, 4x16) occupy identical
// VGPR layouts (V0: K0|K2 per half-wave, V1: K1|K3), so D += A*A accumulates
// the Gram matrix; diag(D) = sum-of-squares of the 16 rows.
// Per-lane tile slice: lane (h*16+r) holds cols {4k+2h, 4k+2h+1}, k=0..15 of
// row r -- which is exactly the A/B operand layout, loaded as float2s.
// ---------------------------------------------------------------------------
__device__ __forceinline__ float tile_inv_norm(const float2 (&a)[16], int lane) {
  v8f c = {0.f, 0.f, 0.f, 0.f, 0.f, 0.f, 0.f, 0.f};
#pragma unroll
  for (int k = 0; k < 16; ++k) {
    v2f av;
    av[0] = a[k].x;
    av[1] = a[k].y;
    c = __builtin_amdgcn_wmma_f32_16x16x4_f32(
        /*neg_a=*/false, av, /*neg_b=*/false, av,
        /*c_mod=*/(short)0, c, /*reuse_a=*/false, /*reuse_b=*/false);
  }
  // diag(M=j)   lives in VGPR j,   lane j        (j = 0..7)
  // diag(M=8+j) lives in VGPR j,   lane 24+j
  int row = lane & 15;
  float ss = 0.f;
#pragma unroll
  for (int j = 0; j < 8; ++j) {
    float dl = __shfl(c[j], j, 32);
    float dh = __shfl(c[j], j + 24, 32);
    ss = (row == j)     ? dl : ss;
    ss = (row == j + 8) ? dh : ss;
  }
  return 1.0f / fmaxf(__builtin_sqrtf(ss), EPSF);
}

// Element offset of this lane's slice of tile `tile` (16 rows of 64 cols).
__device__ __forceinline__ size_t tile_lane_offset(int tile, int lane) {
  int row  = lane & 15;
  int half = lane >> 4;
  return (size_t)(tile * 16 + row) * K_DIM + 2 * half;
}

// ---------------------------------------------------------------------------
// x = l2norm(x) in place; acc += l2norm(x).  One wave per 16 rows.
// ---------------------------------------------------------------------------
__global__ void l2norm_acc_kernel(float* __restrict__ x, float* __restrict__ acc,
                                  int ntiles) {
  int gt = blockIdx.x * blockDim.x + threadIdx.x;
  int wave = gt >> 5;               // uniform across the wave
  if (wave >= ntiles) return;       // whole-wave exit -> EXEC all-1s for WMMA
  int lane = gt & 31;
  size_t off = tile_lane_offset(wave, lane);

  float2 a[16];
#pragma unroll
  for (int k = 0; k < 16; ++k) a[k] = *(const float2*)(x + off + 4 * k);

  float inv = tile_inv_norm(a, lane);

#pragma unroll
  for (int k = 0; k < 16; ++k) {
    float2 v;
    v.x = a[k].x * inv;
    v.y = a[k].y * inv;
    *(float2*)(x + off + 4 * k) = v;
    float2* ap = (float2*)(acc + off + 4 * k);
    float2 av = *ap;
    av.x += v.x;
    av.y += v.y;
    *ap = av;
  }
}

// ---------------------------------------------------------------------------
// Item half of output: out = 0.25*out + l2norm(gis)   (out currently = acc)
// ---------------------------------------------------------------------------
__global__ void final_item_kernel(const float* __restrict__ gis,
                                  float* __restrict__ out_item, int ntiles) {
  int gt = blockIdx.x * blockDim.x + threadIdx.x;
  int wave = gt >> 5;
  if (wave >= ntiles) return;
  int lane = gt & 31;
  size_t off = tile_lane_offset(wave, lane);

  float2 a[16];
#pragma unroll
  for (int k = 0; k < 16; ++k) a[k] = *(const float2*)(gis + off + 4 * k);

  float inv = tile_inv_norm(a, lane);

#pragma unroll
  for (int k = 0; k < 16; ++k) {
    float2* op = (float2*)(out_item + off + 4 * k);
    float2 o = *op;
    o.x = 0.25f * o.x + a[k].x * inv;
    o.y = 0.25f * o.y + a[k].y * inv;
    *op = o;
  }
}

// ---------------------------------------------------------------------------
// User half of output: out *= 0.25  (mean over the 4 embedding stages)
// ---------------------------------------------------------------------------
__global__ void scale_kernel(float* __restrict__ p, float s, int n) {
  int idx = blockIdx.x * blockDim.x + threadIdx.x;
  if (idx < n) p[idx] *= s;
}

}  // namespace

extern "C" void kernel_launch(void* const* d_in, const int* in_sizes, int n_in,
                              void* d_out, int out_size, void* d_ws, size_t ws_size,
                              hipStream_t stream) {
  const float* Gu     = (const float*)d_in[0];
  const float* Gi     = (const float*)d_in[1];
  const float* Gis    = (const float*)d_in[2];
  const float* ii_w   = (const float*)d_in[3];
  const float* ui_w   = (const float*)d_in[4];
  const int*   ii_src = (const int*)d_in[5];
  const int*   ii_dst = (const int*)d_in[6];
  const int*   ui_src = (const int*)d_in[7];
  const int*   ui_dst = (const int*)d_in[8];
  float* out = (float*)d_out;

  const int nE_ii = in_sizes[3];
  const int nE_ui = in_sizes[4];

  constexpr int nUI = (U_N + I_N) * K_DIM;  // 9,600,000 floats
  constexpr int nII = I_N * K_DIM;          // 3,200,000 floats
  float* bufA = (float*)d_ws;
  float* bufB = bufA + nUI;
  float* gisA = bufB + nUI;
  float* gisB = gisA + nII;                  // total ws use: 102.4 MB

  constexpr int ZB = 256, ZG = 2048;         // grid-stride zero

  // --- II path: gisB = spmm(Gis); gisA = spmm(gisB) ---
  zero4_kernel<<<ZG, ZB, 0, stream>>>((float4*)gisB, nII / 4);
  spmm_kernel<<<(nE_ii * 16 + 255) / 256, 256, 0, stream>>>(
      ii_src, ii_dst, ii_w, Gis, gisB, nE_ii);
  zero4_kernel<<<ZG, ZB, 0, stream>>>((float4*)gisA, nII / 4);
  spmm_kernel<<<(nE_ii * 16 + 255) / 256, 256, 0, stream>>>(
      ii_src, ii_dst, ii_w, gisB, gisA, nE_ii);

  // --- ego = concat(Gu, Gi); acc (d_out) = ego ---
  init_concat_kernel<<<(nUI + 255) / 256, 256, 0, stream>>>(Gu, Gi, bufA, out);

  // --- 3 UI layers: next = l2norm(spmm(cur)); acc += next ---
  constexpr int tilesUI = (U_N + I_N) / 16;  // 9375 (exact)
  float* cur = bufA;
  float* nxt = bufB;
  for (int layer = 0; layer < 3; ++layer) {
    zero4_kernel<<<ZG, ZB, 0, stream>>>((float4*)nxt, nUI / 4);
    spmm_kernel<<<(nE_ui * 16 + 255) / 256, 256, 0, stream>>>(
        ui_src, ui_dst, ui_w, cur, nxt, nE_ui);
    l2norm_acc_kernel<<<(tilesUI * 32 + 255) / 256, 256, 0, stream>>>(
        nxt, out, tilesUI);
    float* t = cur; cur = nxt; nxt = t;
  }

  // --- finalize: users = acc/4 ; items = acc/4 + l2norm(gis2) ---
  scale_kernel<<<(U_N * K_DIM + 255) / 256, 256, 0, stream>>>(
      out, 0.25f, U_N * K_DIM);
  constexpr int tilesI = I_N / 16;           // 3125 (exact)
  final_item_kernel<<<(tilesI * 32 + 255) / 256, 256, 0, stream>>>(
      gisA, out + (size_t)U_N * K_DIM, tilesI);
}